// GraphReasoning_29205777613396
// MI455X (gfx1250) — compile-verified
//
#include <hip/hip_runtime.h>
#include <math.h>

typedef float v2f __attribute__((ext_vector_type(2)));
typedef float v8f __attribute__((ext_vector_type(8)));

#define MODE_CM   0   // A[k][m] channel-major (vert_a):  base + dm + k*1024
#define MODE_RM   1   // A[m][k] row-major (feat/hidden): base + dm*K + k
#define MODE_TOKB 2   // gather from vert_b (r=2 unfold)
#define MODE_TOKC 3   // gather from vert_c (r=4 unfold)

__device__ __forceinline__ v8f wmma_f32_k4(v2f a, v2f b, v8f c) {
  // D = A(16x4 f32) x B(4x16 f32) + C(16x16 f32), full fp32 matrix pipe
  return __builtin_amdgcn_wmma_f32_16x16x4_f32(false, a, false, b, (short)0, c,
                                               false, false);
}

// Per-lane A pointer for a 16-row tile starting at absolute row mt0.
// ln = lane&15 (row within tile), koff = (lane>>4)*2 (K sub-offset per ISA layout).
template<int MODE, int K>
__device__ __forceinline__ const float* a_tile_ptr(const float* __restrict__ A,
                                                   int mt0, int ln, int koff,
                                                   int& sk) {
  if constexpr (MODE == MODE_CM) {
    int b = mt0 >> 10, n0 = mt0 & 1023;
    sk = 1024;
    return A + (long)b * K * 1024 + n0 + ln + (long)koff * 1024;
  } else if constexpr (MODE == MODE_RM) {
    sk = 1;
    return A + (long)(mt0 + ln) * K + koff;
  } else if constexpr (MODE == MODE_TOKB) {
    // m = (b*4 + p)*1024 + n ; vert_b (8,128,64,64); tok[b,p,n,c]=vb[b,c,2hn+pr,2wn+pc]
    int b = mt0 >> 12, rem = mt0 & 4095;
    int p = rem >> 10, n0 = rem & 1023;
    int hn = n0 >> 5, wn0 = n0 & 31, pr = p >> 1, pc = p & 1;
    sk = 4096;
    return A + (long)b * 524288 + (long)(2 * hn + pr) * 64 + 2 * (wn0 + ln) + pc
             + (long)koff * 4096;
  } else {
    // m = (b*16 + p)*1024 + n ; vert_c (8,64,128,128); tok[b,p,n,c]=vc[b,c,4hn+pr,4wn+pc]
    int b = mt0 >> 14, rem = mt0 & 16383;
    int p = rem >> 10, n0 = rem & 1023;
    int hn = n0 >> 5, wn0 = n0 & 31, pr = p >> 2, pc = p & 3;
    sk = 16384;
    return A + (long)b * 1048576 + (long)(4 * hn + pr) * 128 + 4 * (wn0 + ln) + pc
             + (long)koff * 16384;
  }
}

// One 16x16 output tile, K-dim reduction via WMMA f32 16x16x4.
// Two accumulators interleaved to break the WMMA->WMMA RAW hazard chain.
template<int K, bool VECA>
__device__ __forceinline__ v8f gemm_tile(const float* __restrict__ Ap, int sk,
                                         const float* __restrict__ Wp) {
  v8f acc0 = {0.f, 0.f, 0.f, 0.f, 0.f, 0.f, 0.f, 0.f};
  v8f acc1 = {0.f, 0.f, 0.f, 0.f, 0.f, 0.f, 0.f, 0.f};
#pragma unroll
  for (int k = 0; k < K; k += 8) {
    v2f a0, a1, b0, b1;
    if constexpr (VECA) {
      a0 = *(const v2f*)(Ap + k);
      a1 = *(const v2f*)(Ap + k + 4);
    } else {
      a0.x = Ap[(long)k * sk];       a0.y = Ap[(long)(k + 1) * sk];
      a1.x = Ap[(long)(k + 4) * sk]; a1.y = Ap[(long)(k + 5) * sk];
    }
    b0 = *(const v2f*)(Wp + k);
    b1 = *(const v2f*)(Wp + k + 4);
    acc0 = wmma_f32_k4(a0, b0, acc0);
    acc1 = wmma_f32_k4(a1, b1, acc1);
  }
  return acc0 + acc1;
}

// Fused 2-layer MLP:  E = (relu(A @ W1^T)) @ W2^T  for TM rows per block.
// Hidden kept in LDS (stride H+4 -> conflict-free ds_load_b64 in phase 2).
// OUTCM=true writes channel-major (B,O,1024) layout (final kernel -> d_out).
template<int MODE, bool OUTCM, int TM, int K, int H, int O>
__global__ void __launch_bounds__(256)
fused_mlp(const float* __restrict__ A, const float* __restrict__ W1,
          const float* __restrict__ W2, float* __restrict__ E) {
  extern __shared__ float hid[];
  constexpr int HP = H + 4;
  constexpr int MT = TM / 16;
  const int tid = threadIdx.x;
  const int wave = tid >> 5, lane = tid & 31;
  const int ln = lane & 15, koff = (lane >> 4) * 2;
  const int mro = (lane >> 4) * 8;          // C/D row offset for lanes 16..31
  const int m0 = blockIdx.x * TM;

  // ---- phase 1: hid[TM][H] = relu(A @ W1^T) ----
  constexpr int NT1 = MT * (H / 16);
  for (int t = wave; t < NT1; t += 8) {
    int mt = t & (MT - 1), o0 = (t / MT) * 16;
    int sk;
    const float* Ap = a_tile_ptr<MODE, K>(A, m0 + mt * 16, ln, koff, sk);
    const float* Wp = W1 + (long)(o0 + ln) * K + koff;
    v8f acc = gemm_tile<K, MODE == MODE_RM>(Ap, sk, Wp);
    int lrow = mt * 16 + mro;
#pragma unroll
    for (int v = 0; v < 8; ++v)
      hid[(lrow + v) * HP + o0 + ln] = fmaxf(acc[v], 0.0f);
  }
  __syncthreads();

  // ---- phase 2: E = hid @ W2^T ----
  constexpr int NT2 = MT * (O / 16);
  for (int t = wave; t < NT2; t += 8) {
    int mt = t & (MT - 1), o0 = (t / MT) * 16;
    const float* Hp = hid + (mt * 16 + ln) * HP + koff;
    const float* Wp = W2 + (long)(o0 + ln) * H + koff;
    v8f acc = gemm_tile<H, true>(Hp, 1, Wp);
    if constexpr (OUTCM) {
      int b  = m0 >> 10;
      int nb = (m0 & 1023) + mt * 16 + mro;
      float* Ep = E + (long)b * (O * 1024) + (long)(o0 + ln) * 1024 + nb;
#pragma unroll
      for (int v = 0; v < 8; ++v) Ep[v] = acc[v];
    } else {
      float* Ep = E + (long)(m0 + mt * 16 + mro) * O + o0 + ln;
#pragma unroll
      for (int v = 0; v < 8; ++v) Ep[(long)v * O] = acc[v];
    }
  }
}

// Single GEMM with gate epilogue: E = 1 - sigmoid(A @ W^T) = 1/(1+exp(x))
template<int K, int O>
__global__ void __launch_bounds__(256)
gemm_gate(const float* __restrict__ A, const float* __restrict__ W,
          float* __restrict__ E) {
  const int tid = threadIdx.x;
  const int wave = tid >> 5, lane = tid & 31;
  const int ln = lane & 15, koff = (lane >> 4) * 2;
  const int mro = (lane >> 4) * 8;
  const int m0 = blockIdx.x * 64;
  constexpr int NT = 4 * (O / 16);
  for (int t = wave; t < NT; t += 8) {
    int mt = t & 3, o0 = (t >> 2) * 16;
    int sk;
    const float* Ap = a_tile_ptr<MODE_CM, K>(A, m0 + mt * 16, ln, koff, sk);
    const float* Wp = W + (long)(o0 + ln) * K + koff;
    v8f acc = gemm_tile<K, false>(Ap, sk, Wp);
    float* Ep = E + (long)(m0 + mt * 16 + mro) * O + o0 + ln;
#pragma unroll
    for (int v = 0; v < 8; ++v) {
      float x = acc[v];
      Ep[(long)v * O] = 1.0f / (1.0f + expf(x));   // 1 - sigmoid(x)
    }
  }
}

// Softmax aggregation for one source (P = 4 or 16); writes sigmoid(gated agg).
template<int P>
__device__ __forceinline__ void agg_one(const float* __restrict__ emb, int b, int n,
                                        float sa, const float* __restrict__ wb,
                                        const float* __restrict__ gate, int m,
                                        float* __restrict__ dst) {
  long roff[P];
  float score[P];
#pragma unroll
  for (int p = 0; p < P; ++p)
    roff[p] = ((long)(b * P + p) * 1024 + n) * 128;
#pragma unroll
  for (int p = 0; p < P; ++p) {
    const float* row = emb + roff[p];
    float s = 0.f;
    for (int c = 0; c < 128; ++c) s += row[c] * wb[c];
    score[p] = fmaxf(sa + s, 0.f);
  }
  float mx = score[0];
#pragma unroll
  for (int p = 1; p < P; ++p) mx = fmaxf(mx, score[p]);
  float denom = 0.f;
#pragma unroll
  for (int p = 0; p < P; ++p) { score[p] = expf(score[p] - mx); denom += score[p]; }
  float inv = 1.f / denom;
#pragma unroll
  for (int p = 0; p < P; ++p) score[p] *= inv;
  const float* g = gate + (long)m * 128;
  for (int c = 0; c < 128; ++c) {
    float acc = 0.f;
#pragma unroll
    for (int p = 0; p < P; ++p) acc += score[p] * emb[roff[p] + c];
    acc *= g[c];
    dst[c] = 1.f / (1.f + expf(-acc));
  }
}

__global__ void __launch_bounds__(256)
aggregate_kernel(const float* __restrict__ embA, const float* __restrict__ embB,
                 const float* __restrict__ embC, const float* __restrict__ gateB,
                 const float* __restrict__ gateC, const float* __restrict__ wab,
                 const float* __restrict__ wac, float* __restrict__ feat) {
  int m = blockIdx.x * blockDim.x + threadIdx.x;   // 0..8191
  int b = m >> 10, n = m & 1023;
  const float* ea = embA + (long)m * 128;
  float sab = 0.f, sac = 0.f;
  for (int c = 0; c < 128; ++c) {
    float e = ea[c];
    sab += e * wab[c];
    sac += e * wac[c];
  }
  float* f = feat + (long)m * 384;
  agg_one<4>(embB, b, n, sab, wab + 128, gateB, m, f);
  agg_one<16>(embC, b, n, sac, wac + 128, gateC, m, f + 128);
  for (int c = 0; c < 128; ++c)
    f[256 + c] = 1.f / (1.f + expf(-ea[c]));
}

extern "C" void kernel_launch(void* const* d_in, const int* in_sizes, int n_in,
                              void* d_out, int out_size, void* d_ws, size_t ws_size,
                              hipStream_t stream) {
  (void)in_sizes; (void)n_in; (void)out_size; (void)ws_size;
  const float* vert_a = (const float*)d_in[0];
  const float* vert_b = (const float*)d_in[1];
  const float* vert_c = (const float*)d_in[2];
  const float* Wa1    = (const float*)d_in[3];
  const float* Wa2    = (const float*)d_in[4];
  const float* Wgb    = (const float*)d_in[5];
  const float* Wgc    = (const float*)d_in[6];
  const float* Wb1    = (const float*)d_in[7];
  const float* Wb2    = (const float*)d_in[8];
  const float* Wc1    = (const float*)d_in[9];
  const float* Wc2    = (const float*)d_in[10];
  const float* wab    = (const float*)d_in[11];
  const float* wac    = (const float*)d_in[12];
  const float* Wr1    = (const float*)d_in[13];
  const float* Wr2    = (const float*)d_in[14];

  // Workspace layout (floats). Total 26M floats = 104 MB (fits the 192 MB L2).
  float* ws    = (float*)d_ws;
  float* embA  = ws;                        //  8192*128 = 1M
  float* gateB = ws + (1u  << 20);          //  1M
  float* gateC = ws + (2u  << 20);          //  1M
  float* embB  = ws + (3u  << 20);          // 32768*128 = 4M
  float* embC  = ws + (7u  << 20);          // 131072*128 = 16M
  float* feat  = ws + (23u << 20);          //  8192*384 = 3M

  dim3 blk(256);
  // Stage A: embA = relu(va@Wa1^T)@Wa2^T   (K=256, H=128, O=128)
  fused_mlp<MODE_CM, false, 64, 256, 128, 128>
      <<<128, blk, 64 * 132 * 4, stream>>>(vert_a, Wa1, Wa2, embA);
  // Gates: 1 - sigmoid(va@Wg^T)
  gemm_gate<256, 128><<<128, blk, 0, stream>>>(vert_a, Wgb, gateB);
  gemm_gate<256, 128><<<128, blk, 0, stream>>>(vert_a, Wgc, gateC);
  // Stage B: embB = relu(tok_b@Wb1^T)@Wb2^T (unfold fused into gather)
  fused_mlp<MODE_TOKB, false, 64, 128, 128, 128>
      <<<512, blk, 64 * 132 * 4, stream>>>(vert_b, Wb1, Wb2, embB);
  // Stage C: embC = relu(tok_c@Wc1^T)@Wc2^T
  fused_mlp<MODE_TOKC, false, 64, 64, 128, 128>
      <<<2048, blk, 64 * 132 * 4, stream>>>(vert_c, Wc1, Wc2, embC);
  // Softmax aggregation + gating + sigmoid -> feat (8192 x 384 row-major)
  aggregate_kernel<<<32, blk, 0, stream>>>(embA, embB, embC, gateB, gateC,
                                           wab, wac, feat);
  // Final: out = relu(feat@Wr1^T)@Wr2^T, channel-major (B,256,32,32).
  // TM=32 keeps LDS (32*260*4 = 33 KB) well under limits with H=256.
  fused_mlp<MODE_RM, true, 32, 384, 256, 256>
      <<<256, blk, 32 * 260 * 4, stream>>>(feat, Wr1, Wr2, (float*)d_out);
}